// Difference3DCostVolume_14310831031098
// MI455X (gfx1250) — compile-verified
//
#include <hip/hip_runtime.h>
#include <cstdint>
#include <cstddef>

// Problem constants from the reference.
#define MAX_DISP 48
#define B_DIM 2
#define C_DIM 32
#define H_DIM 128
#define W_DIM 240

typedef float v4f __attribute__((ext_vector_type(4)));

// out[b,c,d,h,w] = (w >= d) ? l[b,c,h,w] - r[b,c,h,w-d] : 1.0f
//
// One block per (b,c,h) row. Stage the l-row and r-row (240 floats each) into
// LDS with gfx1250 async global->LDS b128 loads (ASYNCcnt), then emit the
// 48 x 240 output slab with 128-bit non-temporal stores (write-once data,
// bigger than L2 -> NT keeps L2 clean).
__global__ __launch_bounds__(256) void
Difference3DCostVolume_kernel(const float* __restrict__ l,
                              const float* __restrict__ r,
                              float* __restrict__ out) {
    constexpr int D  = MAX_DISP;
    constexpr int H  = H_DIM;
    constexpr int W  = W_DIM;       // 240
    constexpr int HW = H * W;       // 30720
    constexpr int Q  = W / 4;       // 60 float4 columns per d-row

    __shared__ __align__(16) float s_rows[2 * W];   // [0..239]=l row, [240..479]=r row

    const int tid = threadIdx.x;
    const int row = blockIdx.x;     // row = bc * H + h, bc = b*C + c
    const int h   = row % H;
    const int bc  = row / H;

    const float* lrow = l + (size_t)bc * HW + (size_t)h * W;
    const float* rrow = r + (size_t)bc * HW + (size_t)h * W;

    // ---- Stage both rows into LDS via async b128 loads (gfx1250 ASYNCcnt path).
    // 120 lanes x 16B = 1920B = 480 floats. Both rows are 16B-aligned
    // (h*W*4 = h*960, bc*HW*4 multiples of 16; hipMalloc base is 256B aligned).
    const uint32_t lds_base =
        (uint32_t)(uintptr_t)((__attribute__((address_space(3))) float*)s_rows);
    if (tid < 2 * Q) {              // lanes 0..119
        const float* g = (tid < Q) ? (lrow + tid * 4)
                                   : (rrow + (tid - Q) * 4);
        const uint32_t lds_addr = lds_base + (uint32_t)tid * 16u;
        asm volatile("global_load_async_to_lds_b128 %0, %1, off"
                     :
                     : "v"(lds_addr), "v"(g)
                     : "memory");
    }
    // Drain this wave's async copies, then make LDS visible to the whole block.
    asm volatile("s_wait_asynccnt 0x0" ::: "memory");
    __syncthreads();

    const float* sL = s_rows;
    const float* sR = s_rows + W;

    // Base of the (b,c,*,h,*) output slab; +d*HW steps the disparity plane.
    float* orow = out + (size_t)bc * D * HW + (size_t)h * W;

    // 48 d-planes * 60 float4 columns = 2880 vec4 stores per block,
    // ~11.25 per thread. Pure streaming: 128-bit non-temporal stores.
    for (int i = tid; i < D * Q; i += 256) {
        const int d  = i / Q;
        const int w0 = (i - d * Q) * 4;
        v4f v;
#pragma unroll
        for (int k = 0; k < 4; ++k) {
            const int w = w0 + k;
            const int j = w - d;
            const float diff = sL[w] - sR[j < 0 ? 0 : j];
            v[k] = (w >= d) ? diff : 1.0f;
        }
        __builtin_nontemporal_store(v, (v4f*)(orow + (size_t)d * HW + w0));
    }
}

extern "C" void kernel_launch(void* const* d_in, const int* in_sizes, int n_in,
                              void* d_out, int out_size, void* d_ws, size_t ws_size,
                              hipStream_t stream) {
    (void)in_sizes; (void)n_in; (void)out_size; (void)d_ws; (void)ws_size;
    const float* l_fmap = (const float*)d_in[0];
    const float* r_fmap = (const float*)d_in[1];
    float* out = (float*)d_out;

    const int rows = B_DIM * C_DIM * H_DIM;   // 8192 blocks, one per (b,c,h)
    Difference3DCostVolume_kernel<<<dim3(rows), dim3(256), 0, stream>>>(l_fmap, r_fmap, out);
}